// MultiRelationalGNN_59742995088126
// MI455X (gfx1250) — compile-verified
//
#include <hip/hip_runtime.h>
#include <hip/hip_bf16.h>
#include <climits>

#define NNODES 50000
#define NFEAT  256
#define NHEAD  4
#define NCHAN  64
#define NEDGE  400000
#define HC     256     // NHEAD*NCHAN
#define HIDW   64

typedef __attribute__((ext_vector_type(2))) float v2f;
typedef __attribute__((ext_vector_type(8))) float v8f;

// Order-preserving float<->int mapping so signed-int atomicMax == float max.
__device__ __forceinline__ int f2ord(float f) {
  int i = __float_as_int(f);
  return i >= 0 ? i : (i ^ 0x7fffffff);
}
__device__ __forceinline__ float ord2f(int i) {
  return __int_as_float(i >= 0 ? i : (i ^ 0x7fffffff));
}

// out[M x Nc] = act( A[M x K] @ W[Nc x K]^T + bias ); act: 0=none, 1=ELU.
// One wave per 16x16 output tile, fp32 WMMA, K stepped by 4.
// A-frag: lane (m = lane&15, khalf = (lane>>4)*2) holds A[m][k+khalf], A[m][k+khalf+1]
// B-frag: same pattern on W rows (output columns), since B = W^T.
// D: VGPR v, lane l -> row = v + 8*(l>>4), col = l&15.
__global__ void gemm_wmma_f32(const float* __restrict__ A,
                              const float* __restrict__ W,
                              const float* __restrict__ bias,
                              float* __restrict__ out,
                              int M, int K, int Nc, int act) {
  const int lane   = threadIdx.x & 31;
  const int wave   = threadIdx.x >> 5;
  const int tilesN = Nc >> 4;
  const int numTiles = (M >> 4) * tilesN;
  const int tile = blockIdx.x * (blockDim.x >> 5) + wave;
  if (tile >= numTiles) return;      // wave-uniform: EXEC stays all-1s for WMMA
  const int tm = tile / tilesN;
  const int tn = tile - tm * tilesN;
  const int r     = lane & 15;
  const int khalf = (lane >> 4) << 1;    // 0 or 2
  const float* arow = A + (size_t)(tm * 16 + r) * K + khalf;
  const float* wrow = W + (size_t)(tn * 16 + r) * K + khalf;
  v8f c = {};
  for (int k = 0; k < K; k += 4) {
    v2f a, b;
    a.x = arow[k];  a.y = arow[k + 1];
    b.x = wrow[k];  b.y = wrow[k + 1];
    c = __builtin_amdgcn_wmma_f32_16x16x4_f32(false, a, false, b,
                                              (short)0, c, false, false);
  }
  const int col   = tn * 16 + r;
  const int rbase = tm * 16 + ((lane >> 4) << 3);
  const float bs = bias ? bias[col] : 0.0f;
#pragma unroll
  for (int v = 0; v < 8; ++v) {
    float val = c[v] + bs;
    if (act == 1) val = val > 0.0f ? val : (__expf(val) - 1.0f);
    out[(size_t)(rbase + v) * Nc + col] = val;
  }
}

// alpha_src[n,h] = sum_c h[n,h,c]*a_src[h,c]; same for dst.
__global__ void alpha_kernel(const float* __restrict__ hbuf,
                             const float* __restrict__ asrc,
                             const float* __restrict__ adst,
                             float* __restrict__ al_s,
                             float* __restrict__ al_d) {
  int t = blockIdx.x * blockDim.x + threadIdx.x;
  if (t >= NNODES * NHEAD) return;
  int n = t >> 2, h = t & 3;
  const float* hp = hbuf + (size_t)n * HC + h * NCHAN;
  const float* as = asrc + h * NCHAN;
  const float* ad = adst + h * NCHAN;
  float ss = 0.f, sd = 0.f;
  for (int cc = 0; cc < NCHAN; ++cc) {
    float hv = hp[cc];
    ss += hv * as[cc];
    sd += hv * ad[cc];
  }
  al_s[t] = ss;
  al_d[t] = sd;
}

__global__ void fill_f32(float* p, float v, int n) {
  int i = blockIdx.x * blockDim.x + threadIdx.x;
  if (i < n) p[i] = v;
}
__global__ void fill_i32(int* p, int v, int n) {
  int i = blockIdx.x * blockDim.x + threadIdx.x;
  if (i < n) p[i] = v;
}

// Pass 1: e = leaky_relu(alpha_src[s]+alpha_dst[d], 0.2); segment-max into mmax[dst].
__global__ void edge_max_kernel(const int* __restrict__ src, const int* __restrict__ dst,
                                const float* __restrict__ al_s, const float* __restrict__ al_d,
                                float* __restrict__ ews, int* __restrict__ mmax) {
  int t = blockIdx.x * blockDim.x + threadIdx.x;
  const int total = (NEDGE + NNODES) * NHEAD;
  if (t >= total) return;
  int i = t >> 2, h = t & 3;
  int s = (i < NEDGE) ? src[i] : (i - NEDGE);  // appended self loops
  int d = (i < NEDGE) ? dst[i] : (i - NEDGE);
  float e = al_s[s * NHEAD + h] + al_d[d * NHEAD + h];
  e = e > 0.f ? e : 0.2f * e;
  ews[t] = e;
  atomicMax(&mmax[d * NHEAD + h], f2ord(e));
}

// Pass 2: p = exp(e - m[dst]); denom[dst] += p; msg[dst,:] += p * h[src,:].
// One 256-thread block per edge: thread t -> (h = t>>6, c = t&63).
__global__ void edge_msg_kernel(const int* __restrict__ src, const int* __restrict__ dst,
                                const float* __restrict__ ews, const int* __restrict__ mmax,
                                const float* __restrict__ hbuf,
                                float* __restrict__ denom, float* __restrict__ msg) {
  int i = blockIdx.x;
  int t = threadIdx.x;
  int h = t >> 6;
  int s = (i < NEDGE) ? src[i] : (i - NEDGE);
  int d = (i < NEDGE) ? dst[i] : (i - NEDGE);
  float e = ews[i * NHEAD + h];
  float m = ord2f(mmax[d * NHEAD + h]);
  float p = __expf(e - m);
  if ((t & 63) == 0) atomicAdd(&denom[d * NHEAD + h], p);
  atomicAdd(&msg[(size_t)d * HC + t], p * hbuf[(size_t)s * HC + t]);
}

// Pass 3: out = ELU(msg/denom + b) written into concat slice rel.
__global__ void finalize_kernel(const float* __restrict__ msg, const float* __restrict__ denom,
                                const float* __restrict__ b, float* __restrict__ hcat, int rel) {
  int idx = blockIdx.x * blockDim.x + threadIdx.x;
  if (idx >= NNODES * HC) return;
  int n = idx >> 8, t = idx & 255, h = t >> 6;
  float v = msg[idx] / denom[n * NHEAD + h] + b[t];
  v = v > 0.f ? v : (__expf(v) - 1.f);
  hcat[(size_t)n * (HC * 3) + rel * HC + t] = v;
}

// fc2: [N,32] -> [N,2] (Nc=2 not WMMA-tileable; trivial flops)
__global__ void fc2_kernel(const float* __restrict__ h1, const float* __restrict__ W,
                           const float* __restrict__ b, float* __restrict__ out) {
  int n = blockIdx.x * blockDim.x + threadIdx.x;
  if (n >= NNODES) return;
  const float* hp = h1 + (size_t)n * 32;
#pragma unroll
  for (int c = 0; c < 2; ++c) {
    float acc = b[c];
#pragma unroll
    for (int j = 0; j < 32; ++j) acc += hp[j] * W[c * 32 + j];
    out[n * 2 + c] = acc;
  }
}

extern "C" void kernel_launch(void* const* d_in, const int* in_sizes, int n_in,
                              void* d_out, int out_size, void* d_ws, size_t ws_size,
                              hipStream_t stream) {
  (void)in_sizes; (void)n_in; (void)out_size; (void)ws_size;
  const float* x = (const float*)d_in[0];
  const int* ei[3]   = {(const int*)d_in[1], (const int*)d_in[2], (const int*)d_in[3]};
  const float* Wr[3]   = {(const float*)d_in[4],  (const float*)d_in[8],  (const float*)d_in[12]};
  const float* asrc[3] = {(const float*)d_in[5],  (const float*)d_in[9],  (const float*)d_in[13]};
  const float* adst[3] = {(const float*)d_in[6],  (const float*)d_in[10], (const float*)d_in[14]};
  const float* br[3]   = {(const float*)d_in[7],  (const float*)d_in[11], (const float*)d_in[15]};
  const float* combine_W = (const float*)d_in[16];
  const float* combine_b = (const float*)d_in[17];
  const float* fc1_W = (const float*)d_in[18];
  const float* fc1_b = (const float*)d_in[19];
  const float* fc2_W = (const float*)d_in[20];
  const float* fc2_b = (const float*)d_in[21];
  float* out = (float*)d_out;

  // Workspace carve (floats)
  float* ws = (float*)d_ws;
  size_t off = 0;
  float* hcat = ws + off; off += (size_t)NNODES * HC * 3;       // [N,768]
  float* hbuf = ws + off; off += (size_t)NNODES * HC;           // [N,256]
  float* msg  = ws + off; off += (size_t)NNODES * HC;           // [N,256]
  float* al_s = ws + off; off += (size_t)NNODES * NHEAD;
  float* al_d = ws + off; off += (size_t)NNODES * NHEAD;
  int*   mmax = (int*)(ws + off); off += (size_t)NNODES * NHEAD;
  float* denom = ws + off; off += (size_t)NNODES * NHEAD;
  float* ews   = ws + off; off += (size_t)(NEDGE + NNODES) * NHEAD;
  float* comb  = ws + off; off += (size_t)NNODES * HIDW;        // [N,64]
  float* h1    = ws + off; off += (size_t)NNODES * (HIDW / 2);  // [N,32]

  const int TPB = 256;
  const int nh = NNODES * NHEAD;
  const int nhc = NNODES * HC;
  const int etot = (NEDGE + NNODES) * NHEAD;

  for (int rel = 0; rel < 3; ++rel) {
    int tiles = (NNODES / 16) * (HC / 16);
    gemm_wmma_f32<<<(tiles + 7) / 8, 256, 0, stream>>>(x, Wr[rel], nullptr, hbuf,
                                                       NNODES, NFEAT, HC, 0);
    alpha_kernel<<<(nh + TPB - 1) / TPB, TPB, 0, stream>>>(hbuf, asrc[rel], adst[rel], al_s, al_d);
    fill_i32<<<(nh + TPB - 1) / TPB, TPB, 0, stream>>>(mmax, INT_MIN, nh);
    fill_f32<<<(nh + TPB - 1) / TPB, TPB, 0, stream>>>(denom, 0.f, nh);
    fill_f32<<<(nhc + TPB - 1) / TPB, TPB, 0, stream>>>(msg, 0.f, nhc);
    edge_max_kernel<<<(etot + TPB - 1) / TPB, TPB, 0, stream>>>(ei[rel], ei[rel] + NEDGE,
                                                                al_s, al_d, ews, mmax);
    edge_msg_kernel<<<NEDGE + NNODES, 256, 0, stream>>>(ei[rel], ei[rel] + NEDGE,
                                                        ews, mmax, hbuf, denom, msg);
    finalize_kernel<<<(nhc + TPB - 1) / TPB, TPB, 0, stream>>>(msg, denom, br[rel], hcat, rel);
  }

  {
    int tiles = (NNODES / 16) * (HIDW / 16);
    gemm_wmma_f32<<<(tiles + 7) / 8, 256, 0, stream>>>(hcat, combine_W, combine_b, comb,
                                                       NNODES, HC * 3, HIDW, 1);
  }
  {
    int tiles = (NNODES / 16) * ((HIDW / 2) / 16);
    gemm_wmma_f32<<<(tiles + 7) / 8, 256, 0, stream>>>(comb, fc1_W, fc1_b, h1,
                                                       NNODES, HIDW, HIDW / 2, 1);
  }
  fc2_kernel<<<(NNODES + TPB - 1) / TPB, TPB, 0, stream>>>(h1, fc2_W, fc2_b, out);
}